// MaskLayer_61684320305653
// MI455X (gfx1250) — compile-verified
//
#include <hip/hip_runtime.h>
#include <cmath>

// MaskLayer: x [B,14,14,512] f32 NHWC.
// Per (b,c): argmax over 196 spatial slots (first occurrence), then
// out = x * TAU * max(1 - BETA*manhattan_dist/14, -1).
//
// Memory-bound (411 MB total traffic -> ~18us floor @ 23.3 TB/s).
// CDNA5 path: stage each (batch, 64-channel) tile global->LDS with
// GLOBAL_LOAD_ASYNC_TO_LDS_B32 (single HBM read), argmax + mask from LDS,
// nontemporal stores for the write-once output.

#define IMG      14
#define SPATIAL  (IMG * IMG)      // 196
#define DEPTH    512
#define BLOCK    64               // threads per block == channels per block (2 waves)
#define CGROUPS  (DEPTH / BLOCK)  // 8

static constexpr float TAU         = 0.5f / (float)(IMG * IMG); // 0.5/196
static constexpr float BETA_OVER_N = 4.0f / (float)IMG;         // 4/14

__global__ __launch_bounds__(BLOCK) void
mask_layer_kernel(const float* __restrict__ x, float* __restrict__ out) {
    // [pos][channel] layout: lane-consecutive channels -> conflict-free banks.
    __shared__ float tile[SPATIAL * BLOCK]; // 50176 B

    const int tid = threadIdx.x;                 // channel within group
    const int blk = blockIdx.x;
    const int b   = blk >> 3;                    // / CGROUPS
    const int cg  = blk & (CGROUPS - 1);

    const size_t base = (size_t)b * (SPATIAL * DEPTH) + (size_t)cg * BLOCK + tid;
    const float* gp = x + base;                  // stride DEPTH between positions
    float*       op = out + base;

    // ---- Stage global -> LDS (read HBM exactly once) -----------------------
#if __has_builtin(__builtin_amdgcn_global_load_async_to_lds_b32)
    #pragma unroll 7
    for (int p = 0; p < SPATIAL; ++p) {
        __builtin_amdgcn_global_load_async_to_lds_b32(
            (__attribute__((address_space(1))) int*)(gp + (size_t)p * DEPTH),
            (__attribute__((address_space(3))) int*)(&tile[p * BLOCK + tid]),
            0, 0);
    }
  #if __has_builtin(__builtin_amdgcn_s_wait_asynccnt)
    __builtin_amdgcn_s_wait_asynccnt(0);
  #else
    asm volatile("s_wait_asynccnt 0" ::: "memory");
  #endif
    asm volatile("" ::: "memory"); // keep LDS reads after the wait
#else
    // Fallback: VGPR-staged nontemporal load + ds_store.
    #pragma unroll 7
    for (int p = 0; p < SPATIAL; ++p)
        tile[p * BLOCK + tid] = __builtin_nontemporal_load(gp + (size_t)p * DEPTH);
#endif

    // ---- Pass 1: first-occurrence argmax over flattened spatial dims -------
    float best = tile[tid];   // p = 0
    int   bi   = 0;
    #pragma unroll 7
    for (int p = 1; p < SPATIAL; ++p) {
        const float v = tile[p * BLOCK + tid];
        if (v > best) { best = v; bi = p; }      // strict > keeps earliest index
    }
    const float imax = (float)(bi / IMG);
    const float jmax = (float)(bi % IMG);

    // ---- Pass 2: mask + scale, streamed out --------------------------------
    int p = 0;
    for (int i = 0; i < IMG; ++i) {
        const float di = fabsf((float)i - imax);
        #pragma unroll
        for (int j = 0; j < IMG; ++j, ++p) {
            const float dist = di + fabsf((float)j - jmax);
            const float m    = TAU * fmaxf(fmaf(-BETA_OVER_N, dist, 1.0f), -1.0f);
            __builtin_nontemporal_store(tile[p * BLOCK + tid] * m,
                                        op + (size_t)p * DEPTH);
        }
    }
}

extern "C" void kernel_launch(void* const* d_in, const int* in_sizes, int n_in,
                              void* d_out, int out_size, void* d_ws, size_t ws_size,
                              hipStream_t stream) {
    const float* x   = (const float*)d_in[0];
    float*       out = (float*)d_out;

    const int B = in_sizes[0] / (SPATIAL * DEPTH); // 512 for the reference shapes
    dim3 grid(B * CGROUPS);
    dim3 block(BLOCK);
    hipLaunchKernelGGL(mask_layer_kernel, grid, block, 0, stream, x, out);
}